// AttnDecoderRnn_75488345194899
// MI455X (gfx1250) — compile-verified
//
#include <hip/hip_runtime.h>
#include <hip/hip_bf16.h>
#include <math.h>

// Model dims (compile-time constants from the reference)
#define Vv 32000
#define Ee 512
#define Hh 1024
#define Ll 50
#define Bb 64
#define BN_EPS 1e-5f

typedef __attribute__((ext_vector_type(2))) float v2f;
typedef __attribute__((ext_vector_type(8))) float v8f;

#define WMMA_F32_4(a, b, c) \
    __builtin_amdgcn_wmma_f32_16x16x4_f32(false, (a), false, (b), (short)0, (c), false, false)

// ---------------------------------------------------------------------------
// Kernel 1: embedding gather + BatchNorm1d (training stats, biased var) + ReLU
// One thread per feature e (E=512). Loops over the B=64 rows.
// ---------------------------------------------------------------------------
__global__ __launch_bounds__(256)
void embed_bn_relu_kernel(const int* __restrict__ ids,
                          const float* __restrict__ emb,
                          const float* __restrict__ gamma,
                          const float* __restrict__ beta,
                          float* __restrict__ x) {
    int e = blockIdx.x * blockDim.x + threadIdx.x;   // e in [0,512)
    float vals[Bb];
    float s = 0.f, s2 = 0.f;
#pragma unroll
    for (int b = 0; b < Bb; ++b) {
        float v = emb[(size_t)ids[b] * Ee + e];
        vals[b] = v;
        s += v;
        s2 += v * v;
    }
    float mu = s * (1.0f / Bb);
    float var = s2 * (1.0f / Bb) - mu * mu;
    float scale = gamma[e] * rsqrtf(var + BN_EPS);
    float shift = beta[e] - mu * scale;
#pragma unroll
    for (int b = 0; b < Bb; ++b) {
        float y = vals[b] * scale + shift;
        x[(size_t)b * Ee + e] = y > 0.f ? y : 0.f;
    }
}

// ---------------------------------------------------------------------------
// WMMA fp32 GEMM core:  C(64 x N) = A(64 x K) * W^T + bias,
// W stored (N,K) row-major (torch Linear layout).  K is COMPILE-TIME so all
// A-row strides / k-offsets become immediate load offsets: one base address
// register per stream, 20 loads per unrolled K-group issued as a deep clause
// into distinct locals, then 16 V_WMMA_F32_16X16X4_F32 ops.
// One wave computes a 64(M) x 16(N) strip; the B (weight) fragment is reused
// from registers across the 4 M-tiles -> weight bytes stream from HBM once.
// Requires: K % 16 == 0, N % (16 * wavesPerBlock) == 0, no divergence.
// ---------------------------------------------------------------------------
template <int K, bool RELU>
__device__ __forceinline__
void wmma_gemm64_core(const float* __restrict__ A,
                      const float* __restrict__ W,
                      const float* __restrict__ bias,
                      float* __restrict__ C,
                      int N, int tn) {
    const int lane  = threadIdx.x & 31;
    const int col   = tn * 16 + (lane & 15);
    const int khalf = (lane >> 4) << 1;                      // 0 or 2

    const float* wp = W + (size_t)col * K + khalf;
    const float* ap = A + (size_t)(lane & 15) * K + khalf;

    v8f acc0 = {}, acc1 = {}, acc2 = {}, acc3 = {};

    for (int k0 = 0; k0 < K; k0 += 16) {
        v2f b[4], a0[4], a1[4], a2[4], a3[4];
#pragma unroll
        for (int u = 0; u < 4; ++u) {
            const int k = k0 + 4 * u;
            b[u]  = *(const v2f*)(wp + k);
            a0[u] = *(const v2f*)(ap + k);
            a1[u] = *(const v2f*)(ap + 16 * K + k);
            a2[u] = *(const v2f*)(ap + 32 * K + k);
            a3[u] = *(const v2f*)(ap + 48 * K + k);
        }
#pragma unroll
        for (int u = 0; u < 4; ++u) {
            acc0 = WMMA_F32_4(a0[u], b[u], acc0);
            acc1 = WMMA_F32_4(a1[u], b[u], acc1);
            acc2 = WMMA_F32_4(a2[u], b[u], acc2);
            acc3 = WMMA_F32_4(a3[u], b[u], acc3);
        }
    }

    const float bv = bias[col];
    const int rowbase = (lane >> 4) ? 8 : 0;     // C/D layout: VGPR i -> row rowbase+i
    float* cp = C + col;
#pragma unroll
    for (int i = 0; i < 8; ++i) {
        float v0 = acc0[i] + bv;
        float v1 = acc1[i] + bv;
        float v2 = acc2[i] + bv;
        float v3 = acc3[i] + bv;
        if (RELU) {
            v0 = v0 > 0.f ? v0 : 0.f;
            v1 = v1 > 0.f ? v1 : 0.f;
            v2 = v2 > 0.f ? v2 : 0.f;
            v3 = v3 > 0.f ? v3 : 0.f;
        }
        int r = rowbase + i;
        cp[(size_t)(r)      * N] = v0;
        cp[(size_t)(r + 16) * N] = v1;
        cp[(size_t)(r + 32) * N] = v2;
        cp[(size_t)(r + 48) * N] = v3;
    }
}

// Standalone GEMM (used for lin and logits)
template <int K, bool RELU>
__global__ __launch_bounds__(256)
void wmma_gemm64_kernel(const float* __restrict__ A,
                        const float* __restrict__ W,
                        const float* __restrict__ bias,
                        float* __restrict__ C,
                        int N) {
    const int tn = blockIdx.x * (blockDim.x >> 5) + (threadIdx.x >> 5);
    wmma_gemm64_core<K, RELU>(A, W, bias, C, N, tn);
}

// Batched GEMM: 4 independent slices (gi_f, gh_f, gi_b, gh_b of one GRU layer)
// launched concurrently via gridDim.y. All slices share N; per-slice K is a
// template constant selected by a block-uniform switch (no lane divergence,
// EXEC stays all-1s inside each slice's GEMM body).
struct GemmSlice {
    const float* A;
    const float* W;
    const float* bias;
    float*       C;
};
struct GemmBatch4 {
    GemmSlice s[4];
};

template <int K0, int K1, int K2, int K3>
__global__ __launch_bounds__(256)
void wmma_gemm64_batch4_kernel(GemmBatch4 batch, int N) {
    const int tn = blockIdx.x * (blockDim.x >> 5) + (threadIdx.x >> 5);
    const GemmSlice sl = batch.s[blockIdx.y];
    switch (blockIdx.y) {
        case 0:  wmma_gemm64_core<K0, false>(sl.A, sl.W, sl.bias, sl.C, N, tn); break;
        case 1:  wmma_gemm64_core<K1, false>(sl.A, sl.W, sl.bias, sl.C, N, tn); break;
        case 2:  wmma_gemm64_core<K2, false>(sl.A, sl.W, sl.bias, sl.C, N, tn); break;
        default: wmma_gemm64_core<K3, false>(sl.A, sl.W, sl.bias, sl.C, N, tn); break;
    }
}

// ---------------------------------------------------------------------------
// GRU gate nonlinearity (torch gate order r,z,n):
//   r = sigmoid(gi_r + gh_r); z = sigmoid(gi_z + gh_z)
//   n = tanh(gi_n + r * gh_n); h' = (1-z)*n + z*h
// Writes h' to the hidden_out slot and into a (B,2H) concat buffer.
// ---------------------------------------------------------------------------
__global__ __launch_bounds__(256)
void gru_gate_kernel(const float* __restrict__ gi,
                     const float* __restrict__ gh,
                     const float* __restrict__ hprev,
                     float* __restrict__ hout,
                     float* __restrict__ concat_out,
                     int coloff) {
    int idx = blockIdx.x * blockDim.x + threadIdx.x;   // B*H
    int b = idx >> 10;            // /H
    int j = idx & (Hh - 1);
    const float* gib = gi + (size_t)b * 3 * Hh;
    const float* ghb = gh + (size_t)b * 3 * Hh;
    float r = 1.f / (1.f + __expf(-(gib[j] + ghb[j])));
    float z = 1.f / (1.f + __expf(-(gib[Hh + j] + ghb[Hh + j])));
    float n = tanhf(gib[2 * Hh + j] + r * ghb[2 * Hh + j]);
    float h = hprev[idx];
    float hn = (1.f - z) * n + z * h;
    hout[idx] = hn;
    concat_out[(size_t)b * (2 * Hh) + coloff + j] = hn;
}

// ---------------------------------------------------------------------------
// Dot-product attention: scores[l,b] = <enc[l,b,:], g[b,:]>, softmax over l,
// attn[b,:] = sum_l w[l,b] * enc[l,b,:].  One block per batch row b.
// ---------------------------------------------------------------------------
__global__ __launch_bounds__(256)
void attention_kernel(const float* __restrict__ enc,
                      const float* __restrict__ g,
                      float* __restrict__ attn) {
    __shared__ float red[256];
    __shared__ float wgt[64];
    const int b = blockIdx.x;
    const int tid = threadIdx.x;
    const float* gb = g + (size_t)b * (2 * Hh);

    for (int l = 0; l < Ll; ++l) {
        const float* e = enc + ((size_t)l * Bb + b) * (2 * Hh);
        float p = 0.f;
        for (int d = tid; d < 2 * Hh; d += 256) p += e[d] * gb[d];
        red[tid] = p;
        __syncthreads();
        for (int s = 128; s > 0; s >>= 1) {
            if (tid < s) red[tid] += red[tid + s];
            __syncthreads();
        }
        if (tid == 0) wgt[l] = red[0];
        __syncthreads();
    }
    if (tid == 0) {
        float m = -INFINITY;
        for (int l = 0; l < Ll; ++l) m = fmaxf(m, wgt[l]);
        float s = 0.f;
        for (int l = 0; l < Ll; ++l) { float v = __expf(wgt[l] - m); wgt[l] = v; s += v; }
        float inv = 1.f / s;
        for (int l = 0; l < Ll; ++l) wgt[l] *= inv;
    }
    __syncthreads();
    for (int d = tid; d < 2 * Hh; d += 256) {
        float a = 0.f;
        for (int l = 0; l < Ll; ++l)
            a += wgt[l] * enc[((size_t)l * Bb + b) * (2 * Hh) + d];
        attn[(size_t)b * (2 * Hh) + d] = a;
    }
}

// ---------------------------------------------------------------------------
// Row-wise log_softmax over V=32000. One block per batch row.
// ---------------------------------------------------------------------------
__global__ __launch_bounds__(256)
void log_softmax_kernel(const float* __restrict__ logits,
                        float* __restrict__ out) {
    __shared__ float red[256];
    const int b = blockIdx.x;
    const int tid = threadIdx.x;
    const float* p = logits + (size_t)b * Vv;

    float m = -INFINITY;
    for (int v = tid; v < Vv; v += 256) m = fmaxf(m, p[v]);
    red[tid] = m;
    __syncthreads();
    for (int s = 128; s > 0; s >>= 1) {
        if (tid < s) red[tid] = fmaxf(red[tid], red[tid + s]);
        __syncthreads();
    }
    const float mx = red[0];
    __syncthreads();

    float sum = 0.f;
    for (int v = tid; v < Vv; v += 256) sum += __expf(p[v] - mx);
    red[tid] = sum;
    __syncthreads();
    for (int s = 128; s > 0; s >>= 1) {
        if (tid < s) red[tid] += red[tid + s];
        __syncthreads();
    }
    const float lse = logf(red[0]);
    __syncthreads();

    float* ob = out + (size_t)b * Vv;
    for (int v = tid; v < Vv; v += 256) ob[v] = p[v] - mx - lse;
}

// ---------------------------------------------------------------------------
// Host-side launch orchestration
// ---------------------------------------------------------------------------
extern "C" void kernel_launch(void* const* d_in, const int* in_sizes, int n_in,
                              void* d_out, int out_size, void* d_ws, size_t ws_size,
                              hipStream_t stream) {
    (void)in_sizes; (void)n_in; (void)out_size; (void)ws_size;

    const int*   ids    = (const int*)  d_in[0];
    const float* hidden = (const float*)d_in[2];   // (4,B,H)
    const float* enc    = (const float*)d_in[3];   // (L,B,2H)
    const float* emb    = (const float*)d_in[4];   // (V,E)
    const float* gamma  = (const float*)d_in[5];
    const float* beta   = (const float*)d_in[6];
    const float* lin_w  = (const float*)d_in[7];   // (H,2H)
    const float* lin_b  = (const float*)d_in[8];
    const float* out_w  = (const float*)d_in[9];   // (V,H)
    const float* out_b  = (const float*)d_in[10];
    // GRU params: [11..14]=l0f, [15..18]=l0b, [19..22]=l1f, [23..26]=l1b
    const float* w_ih_l0f = (const float*)d_in[11];
    const float* w_hh_l0f = (const float*)d_in[12];
    const float* b_ih_l0f = (const float*)d_in[13];
    const float* b_hh_l0f = (const float*)d_in[14];
    const float* w_ih_l0b = (const float*)d_in[15];
    const float* w_hh_l0b = (const float*)d_in[16];
    const float* b_ih_l0b = (const float*)d_in[17];
    const float* b_hh_l0b = (const float*)d_in[18];
    const float* w_ih_l1f = (const float*)d_in[19];
    const float* w_hh_l1f = (const float*)d_in[20];
    const float* b_ih_l1f = (const float*)d_in[21];
    const float* b_hh_l1f = (const float*)d_in[22];
    const float* w_ih_l1b = (const float*)d_in[23];
    const float* w_hh_l1b = (const float*)d_in[24];
    const float* b_ih_l1b = (const float*)d_in[25];
    const float* b_hh_l1b = (const float*)d_in[26];

    // Workspace layout (floats)
    float* ws      = (float*)d_ws;
    float* x       = ws;                 // (B,E)          32768
    float* x1      = ws + 32768;         // (B,2H)        131072
    float* x2      = ws + 163840;        // (B,2H)        131072
    float* gi_f    = ws + 294912;        // (B,3H)        196608
    float* gh_f    = ws + 491520;        // (B,3H)        196608
    float* gi_b    = ws + 688128;        // (B,3H)        196608
    float* gh_b    = ws + 884736;        // (B,3H)        196608
    float* attn    = ws + 1081344;       // (B,2H)        131072
    float* lin     = ws + 1212416;       // (B,H)          65536
    float* logits  = ws + 1277952;       // (B,V)        2048000

    float* out     = (float*)d_out;      // decoder_output (1,B,V)
    float* hid_out = out + (size_t)Bb * Vv;  // hidden_out (4,B,H)

    const float* h0 = hidden;
    const float* h1 = hidden + 1 * Bb * Hh;
    const float* h2 = hidden + 2 * Bb * Hh;
    const float* h3 = hidden + 3 * Bb * Hh;

    // 1) embedding + BN + relu
    embed_bn_relu_kernel<<<Ee / 256, 256, 0, stream>>>(ids, emb, gamma, beta, x);

    // 2) GRU layer 0: all 4 gate GEMMs in one batched launch (96 blocks)
    {
        GemmBatch4 b4;
        b4.s[0] = { x,  w_ih_l0f, b_ih_l0f, gi_f };
        b4.s[1] = { h0, w_hh_l0f, b_hh_l0f, gh_f };
        b4.s[2] = { x,  w_ih_l0b, b_ih_l0b, gi_b };
        b4.s[3] = { h1, w_hh_l0b, b_hh_l0b, gh_b };
        wmma_gemm64_batch4_kernel<Ee, Hh, Ee, Hh>
            <<<dim3(24, 4), 256, 0, stream>>>(b4, 3 * Hh);
    }

    // 3) GRU layer 0 nonlinearity -> hidden_out[0..1] and x1 = concat(h0f,h0b)
    gru_gate_kernel<<<(Bb * Hh) / 256, 256, 0, stream>>>(gi_f, gh_f, h0, hid_out,            x1, 0);
    gru_gate_kernel<<<(Bb * Hh) / 256, 256, 0, stream>>>(gi_b, gh_b, h1, hid_out + Bb * Hh,  x1, Hh);

    // 4) GRU layer 1: all 4 gate GEMMs in one batched launch
    {
        GemmBatch4 b4;
        b4.s[0] = { x1, w_ih_l1f, b_ih_l1f, gi_f };
        b4.s[1] = { h2, w_hh_l1f, b_hh_l1f, gh_f };
        b4.s[2] = { x1, w_ih_l1b, b_ih_l1b, gi_b };
        b4.s[3] = { h3, w_hh_l1b, b_hh_l1b, gh_b };
        wmma_gemm64_batch4_kernel<2 * Hh, Hh, 2 * Hh, Hh>
            <<<dim3(24, 4), 256, 0, stream>>>(b4, 3 * Hh);
    }

    // 5) GRU layer 1 nonlinearity -> hidden_out[2..3] and x2 = gru_out
    gru_gate_kernel<<<(Bb * Hh) / 256, 256, 0, stream>>>(gi_f, gh_f, h2, hid_out + 2 * Bb * Hh, x2, 0);
    gru_gate_kernel<<<(Bb * Hh) / 256, 256, 0, stream>>>(gi_b, gh_b, h3, hid_out + 3 * Bb * Hh, x2, Hh);

    // 6) attention over encoder positions
    attention_kernel<<<Bb, 256, 0, stream>>>(enc, x2, attn);

    // 7) lin = relu(attn @ lin_w^T + lin_b)   (N = 1024 -> 8 blocks)
    wmma_gemm64_kernel<2 * Hh, true><<<8, 256, 0, stream>>>(attn, lin_w, lin_b, lin, Hh);

    // 8) logits = lin @ out_w^T + out_b       (N = 32000 -> 250 blocks)
    wmma_gemm64_kernel<Hh, false><<<250, 256, 0, stream>>>(lin, out_w, out_b, logits, Vv);

    // 9) log_softmax -> decoder_output
    log_softmax_kernel<<<Bb, 256, 0, stream>>>(logits, out);
}